// DataEmbedding_Stats_61744449848197
// MI455X (gfx1250) — compile-verified
//
#include <hip/hip_runtime.h>

// ---------------------------------------------------------------------------
// Fused rolling-stats + circular conv1d(k=3, 35->512) as a WMMA f32 GEMM.
//   out[b,s,o] = sum_{t=0..2, c=0..34} feats[b,(s-1+t) mod S, c] * W[o,c,t] + bias[o]
//   feats = [x, mean24, max24, min24, std24]  (window edge-replicated at s<0)
// GEMM view: M = B*S = 131072, N = 512, K = 105 (pad 108), A built in LDS.
// Matrix op: V_WMMA_F32_16X16X4_F32 (exact f32 math on the matrix pipe).
// B tile stored n-major so each B fragment is ONE ds_load_b64 (k0,k0+1 adjacent).
// ---------------------------------------------------------------------------

typedef __attribute__((ext_vector_type(2))) float v2f;
typedef __attribute__((ext_vector_type(8))) float v8f;

#define B_      32
#define S_      4096
#define CIN     7
#define KK      105      // 35 features * 3 taps
#define KPAD    108      // 27 chunks of 4
#define DM      512
#define WINDOW  24
#define MTILE   128      // s-rows per block
#define NTILE   64       // out-channels per block
#define FSTRIDE 36       // feats LDS row stride (floats)
#define BSTRIDE 110      // weight LDS row stride (floats, even -> 8B-aligned b64)

__global__ __launch_bounds__(256)
void fused_stats_conv_wmma(const float* __restrict__ x,
                           const float* __restrict__ W,
                           const float* __restrict__ bias,
                           float* __restrict__ out)
{
    __shared__ float feats[(MTILE + 2) * FSTRIDE];   // rows s0-1 .. s0+128
    __shared__ float Bl[NTILE * BSTRIDE];            // W tile, n-major, K zero-padded

    const int tid = threadIdx.x;
    const int nb  = blockIdx.x & 7;          // 512/64 = 8 n-blocks
    const int mb  = blockIdx.x >> 3;         // 1024 m-blocks
    const int b   = mb >> 5;                 // 4096/128 = 32 s-tiles per batch
    const int s0  = (mb & 31) * MTILE;
    const int n0  = nb * NTILE;

    // ---- Stage weights: Bl[n][k] = W[(n0+n)*105 + k], k=105..107 zeroed ----
    // Consecutive threads take consecutive k: coalesced global reads and
    // conflict-free LDS stores (bank = (110*n + k) % 64).
    for (int idx = tid; idx < NTILE * KPAD; idx += 256) {
        int n = idx / KPAD;
        int k = idx - n * KPAD;
        Bl[n * BSTRIDE + k] = (k < KK) ? W[(n0 + n) * KK + k] : 0.0f;
    }

    // ---- Stage rolling stats for rows s0-1 .. s0+128 (circular in s) ----
    for (int task = tid; task < (MTILE + 2) * CIN; task += 256) {
        int r = task / CIN;
        int c = task - r * CIN;
        int sg = s0 - 1 + r;
        if (sg < 0)    sg += S_;
        if (sg >= S_)  sg -= S_;
        float vals[WINDOW];
        float sum = 0.0f, mx = -3.4e38f, mn = 3.4e38f;
        #pragma unroll
        for (int j = 0; j < WINDOW; ++j) {
            int sj = sg - (WINDOW - 1) + j;
            sj = sj < 0 ? 0 : sj;                     // edge replication
            float v = x[(b * S_ + sj) * CIN + c];
            vals[j] = v;
            sum += v;
            mx = fmaxf(mx, v);
            mn = fminf(mn, v);
        }
        float mean = sum * (1.0f / WINDOW);
        float var = 0.0f;
        #pragma unroll
        for (int j = 0; j < WINDOW; ++j) {
            float d = vals[j] - mean;
            var += d * d;
        }
        var *= (1.0f / (WINDOW - 1));
        float sd = sqrtf(var + 1e-12f);
        float* fr = &feats[r * FSTRIDE];
        fr[c]           = vals[WINDOW - 1];  // raw x at sg
        fr[CIN + c]     = mean;
        fr[2 * CIN + c] = mx;
        fr[3 * CIN + c] = mn;
        fr[4 * CIN + c] = sd;
    }
    __syncthreads();

    // ---- WMMA main loop: wave w owns M-subtile w, all 4 N-subtiles ----
    const int wave = tid >> 5;
    const int lane = tid & 31;
    const int l16  = lane & 15;
    const int half = lane >> 4;          // K pair select per ISA A/B layout
    const int mA   = wave * 16 + l16;    // A-row == local feats row base (offset by t)

    v8f acc0 = {}, acc1 = {}, acc2 = {}, acc3 = {};

    // k for vector component j: k = 4*kc + 2*half + j. Track (c = k/3, t = k%3)
    // incrementally (k += 4  =>  c += 1, t += 1, carry at t==3) to avoid div/mod.
    int c0 = 0,        t0 = 2 * half;    // j = 0
    int c1 = half,     t1 = 1 - half;    // j = 1
    // B fragment base: lane reads Bl[(ni*16 + l16)*BSTRIDE + k0 .. k0+1] as b64.
    const float* pb = &Bl[l16 * BSTRIDE + 2 * half];

    #pragma unroll 2
    for (int kc = 0; kc < 26; ++kc) {    // k = 0..103, guard-free
        v2f a;
        a.x = feats[(mA + t0) * FSTRIDE + c0];
        a.y = feats[(mA + t1) * FSTRIDE + c1];
        v2f b0 = *(const v2f*)(pb);
        v2f b1 = *(const v2f*)(pb + 16 * BSTRIDE);
        v2f b2 = *(const v2f*)(pb + 32 * BSTRIDE);
        v2f b3 = *(const v2f*)(pb + 48 * BSTRIDE);
        acc0 = __builtin_amdgcn_wmma_f32_16x16x4_f32(false, a, false, b0, (short)0, acc0, false, false);
        acc1 = __builtin_amdgcn_wmma_f32_16x16x4_f32(false, a, false, b1, (short)0, acc1, false, false);
        acc2 = __builtin_amdgcn_wmma_f32_16x16x4_f32(false, a, false, b2, (short)0, acc2, false, false);
        acc3 = __builtin_amdgcn_wmma_f32_16x16x4_f32(false, a, false, b3, (short)0, acc3, false, false);
        // advance k by 4
        t0 += 1; c0 += 1; if (t0 >= 3) { t0 -= 3; c0 += 1; }
        t1 += 1; c1 += 1; if (t1 >= 3) { t1 -= 3; c1 += 1; }
        pb += 4;
    }

    // tail chunk kc = 26: k = 104..107; only k = 104 is real for A
    // (k=104 -> c=34, t=2 on half 0, j 0); Bl cols 105..107 already zero.
    {
        v2f a;
        a.x = (half == 0) ? feats[(mA + 2) * FSTRIDE + 34] : 0.0f;
        a.y = 0.0f;
        v2f b0 = *(const v2f*)(pb);
        v2f b1 = *(const v2f*)(pb + 16 * BSTRIDE);
        v2f b2 = *(const v2f*)(pb + 32 * BSTRIDE);
        v2f b3 = *(const v2f*)(pb + 48 * BSTRIDE);
        acc0 = __builtin_amdgcn_wmma_f32_16x16x4_f32(false, a, false, b0, (short)0, acc0, false, false);
        acc1 = __builtin_amdgcn_wmma_f32_16x16x4_f32(false, a, false, b1, (short)0, acc1, false, false);
        acc2 = __builtin_amdgcn_wmma_f32_16x16x4_f32(false, a, false, b2, (short)0, acc2, false, false);
        acc3 = __builtin_amdgcn_wmma_f32_16x16x4_f32(false, a, false, b3, (short)0, acc3, false, false);
    }

    // ---- Epilogue: C/D layout -> VGPR r holds M = r (lanes 0-15), r+8 (16-31) ----
    const int sbase = s0 + wave * 16 + 8 * half;
    const float bia0 = bias[n0 + l16];
    const float bia1 = bias[n0 + 16 + l16];
    const float bia2 = bias[n0 + 32 + l16];
    const float bia3 = bias[n0 + 48 + l16];
    #pragma unroll
    for (int r = 0; r < 8; ++r) {
        int s = sbase + r;
        float* o = &out[((size_t)(b * S_ + s)) * DM + n0 + l16];
        o[0]  = acc0[r] + bia0;
        o[16] = acc1[r] + bia1;
        o[32] = acc2[r] + bia2;
        o[48] = acc3[r] + bia3;
    }
}

extern "C" void kernel_launch(void* const* d_in, const int* in_sizes, int n_in,
                              void* d_out, int out_size, void* d_ws, size_t ws_size,
                              hipStream_t stream) {
    const float* x    = (const float*)d_in[0];
    // d_in[1] = x_mark: unused by the reference computation.
    const float* W    = (const float*)d_in[2];   // (512, 35, 3) row-major = (512, 105)
    const float* bias = (const float*)d_in[3];   // (512,)
    float* out = (float*)d_out;                  // (32, 4096, 512) f32

    const int grid = (B_ * S_ / MTILE) * (DM / NTILE);  // 1024 * 8 = 8192
    fused_stats_conv_wmma<<<grid, 256, 0, stream>>>(x, W, bias, out);
}